// DemLocDecoder_13211319402659
// MI455X (gfx1250) — compile-verified
//
#include <hip/hip_runtime.h>
#include <hip/hip_bf16.h>

typedef __attribute__((ext_vector_type(2))) float v2f;
typedef __attribute__((ext_vector_type(8))) float v8f;

#define N_NODES 19
#define N_T     4096
#define LATENT  512
#define HID     2048
#define N_EDGES 342
#define PAD_ROWS 32

// ---- workspace layout (in floats) ----
#define WS_H0    0                              // 32 x 512
#define WS_T1    (WS_H0 + PAD_ROWS * LATENT)    // 32 x 2048
#define WS_H1    (WS_T1 + PAD_ROWS * HID)       // 32 x 2048
#define WS_H1AGG (WS_H1 + PAD_ROWS * HID)       // 32 x 2048
#define WS_T2    (WS_H1AGG + PAD_ROWS * HID)    // 32 x 4096
#define WS_Y     (WS_T2 + PAD_ROWS * N_T)       // 512
#define WS_TOTAL (WS_Y + LATENT)

// ---------------- zero init ----------------
__global__ void zero_kernel(float* __restrict__ p, int n) {
    int i = blockIdx.x * blockDim.x + threadIdx.x;
    int stride = gridDim.x * blockDim.x;
    for (; i < n; i += stride) p[i] = 0.0f;
}

// ---------------- GIN aggregation: out[v] = h[v] + sum_{e: dst==v} h[src[e]] ----------------
__global__ void gin_agg_kernel(const float* __restrict__ h, const int* __restrict__ edges,
                               float* __restrict__ out, int F) {
    int idx = blockIdx.x * blockDim.x + threadIdx.x;
    if (idx >= N_NODES * F) return;
    int node = idx / F;
    int f = idx - node * F;
    const int* src = edges;
    const int* dst = edges + N_EDGES;
    float v = h[(size_t)node * F + f];
    for (int e = 0; e < N_EDGES; ++e) {
        if (dst[e] == node) v += h[(size_t)src[e] * F + f];
    }
    out[(size_t)node * F + f] = v;
}

// ---------------- fp32 WMMA GEMM with split-K atomic accumulation ----------------
// A: padded activations [32 x K] (rows >= 19 are zero), row-major stride K
// W: weights [K x N] row-major
// Out: [19 x N] accumulator (pre-zeroed); partial sums added atomically
// All dims compile-time so B-row offsets fold into load immediate offsets.
template <int K, int N, int KCHUNK>
__global__ __launch_bounds__(128) void gemm_wmma_kernel(
        const float* __restrict__ A, const float* __restrict__ W,
        float* __restrict__ Out) {
    const int lane = threadIdx.x;        // 0..31
    const int half = lane >> 4;          // 0 or 1
    const int lm   = lane & 15;
    const int tile = blockIdx.x * 4 + threadIdx.y;
    const int n    = tile * 16 + lm;     // this lane's output column (B/C/D layout)
    const int k0   = blockIdx.y * KCHUNK;

    // A lane layout (16x4 f32): lane holds row M=lm, K pair base = 2*half
    const float* a0p  = A + (size_t)lm * K        + k0 + 2 * half;  // rows 0..15
    const float* a1p  = A + (size_t)(16 + lm) * K + k0 + 2 * half;  // rows 16..31 (pad zero)
    // B lane layout (4x16 f32): lane holds col N=n, rows kb, kb+1
    const float* bptr = W + (size_t)(k0 + 2 * half) * N + n;

    v8f c0 = {};  // rows 0..15
    v8f c1 = {};  // rows 16..31

    #pragma unroll 4
    for (int kk = 0; kk < KCHUNK; kk += 4) {
        v2f a0 = *(const v2f*)a0p;
        v2f a1 = *(const v2f*)a1p;
        v2f b;
        b.x = bptr[0];
        b.y = bptr[N];
        // D = A*B + C, fp32 16x16x4
        c0 = __builtin_amdgcn_wmma_f32_16x16x4_f32(false, a0, false, b, (short)0, c0, false, false);
        c1 = __builtin_amdgcn_wmma_f32_16x16x4_f32(false, a1, false, b, (short)0, c1, false, false);
        a0p += 4;
        a1p += 4;
        bptr += (size_t)4 * N;
    }

    // C/D layout: VGPR i -> row i + 8*half; second tile holds rows 16..31 (only 16..18 valid)
    #pragma unroll
    for (int i = 0; i < 8; ++i) {
        int r0 = i + 8 * half;
        atomicAdd(&Out[(size_t)r0 * N + n], c0[i]);
        int r1 = 16 + r0;
        if (r1 < N_NODES) atomicAdd(&Out[(size_t)r1 * N + n], c1[i]);
    }
}

// ---------------- bias (+optional ReLU) epilogue, in place over [19 x N] ----------------
__global__ void bias_act_kernel(float* __restrict__ buf, const float* __restrict__ bias,
                                int N, int do_relu) {
    int idx = blockIdx.x * blockDim.x + threadIdx.x;
    if (idx >= N_NODES * N) return;
    int col = idx % N;
    float v = buf[idx] + bias[col];
    if (do_relu) v = v > 0.0f ? v : 0.0f;
    buf[idx] = v;
}

// ---------------- classifier matvec: y[j] += sum_r x[r] * Wc1[r,j], j in [0,512) ----------------
__global__ __launch_bounds__(512) void matvec_kernel(
        const float* __restrict__ x, const float* __restrict__ Wc1,
        float* __restrict__ y, int rows, int rowsPerBlock) {
    int j = threadIdx.x;  // 0..511
    int r0 = blockIdx.x * rowsPerBlock;
    int r1 = r0 + rowsPerBlock;
    if (r1 > rows) r1 = rows;
    float acc = 0.0f;
    for (int r = r0; r < r1; ++r) {
        acc += x[r] * Wc1[(size_t)r * LATENT + j];
    }
    atomicAdd(&y[j], acc);
}

// ---------------- final: sigmoid((y + bc1) . Wc2 + bc2) ----------------
__global__ __launch_bounds__(512) void final_kernel(
        const float* __restrict__ y, const float* __restrict__ bc1,
        const float* __restrict__ Wc2, const float* __restrict__ bc2,
        float* __restrict__ out0) {
    __shared__ float red[512];
    int j = threadIdx.x;
    red[j] = (y[j] + bc1[j]) * Wc2[j];
    __syncthreads();
    for (int s = 256; s > 0; s >>= 1) {
        if (j < s) red[j] += red[j + s];
        __syncthreads();
    }
    if (j == 0) {
        float v = red[0] + bc2[0];
        out0[0] = 1.0f / (1.0f + __expf(-v));
    }
}

extern "C" void kernel_launch(void* const* d_in, const int* in_sizes, int n_in,
                              void* d_out, int out_size, void* d_ws, size_t ws_size,
                              hipStream_t stream) {
    const float* z    = (const float*)d_in[0];
    const int*   eidx = (const int*)  d_in[1];   // edge_idx [2, 342]
    const float* W1a  = (const float*)d_in[2];
    const float* b1a  = (const float*)d_in[3];
    const float* W1b  = (const float*)d_in[4];
    const float* b1b  = (const float*)d_in[5];
    const float* W2a  = (const float*)d_in[6];
    const float* b2a  = (const float*)d_in[7];
    const float* W2b  = (const float*)d_in[8];
    const float* b2b  = (const float*)d_in[9];
    const float* Wc1  = (const float*)d_in[10];
    const float* bc1  = (const float*)d_in[11];
    const float* Wc2  = (const float*)d_in[12];
    const float* bc2  = (const float*)d_in[13];

    float* out   = (float*)d_out;          // out[0] = dem_pred, out[1..] = recon [19,4096]
    float* ws    = (float*)d_ws;
    float* h0    = ws + WS_H0;
    float* t1    = ws + WS_T1;
    float* h1    = ws + WS_H1;
    float* h1agg = ws + WS_H1AGG;
    float* t2    = ws + WS_T2;
    float* yv    = ws + WS_Y;
    float* recon = out + 1;                // [19 x 4096], also "flat" classifier input

    // 0) zero all accumulators + output (fresh every call)
    zero_kernel<<<512, 256, 0, stream>>>(ws, (int)WS_TOTAL);
    zero_kernel<<<512, 256, 0, stream>>>(out, 1 + N_NODES * N_T);

    dim3 gblk(32, 4);

    // 1) agg1: h0 = z + segsum(z)       [19 x 512] into padded [32 x 512]
    {
        int n = N_NODES * LATENT;
        gin_agg_kernel<<<(n + 255) / 256, 256, 0, stream>>>(z, eidx, h0, LATENT);
    }
    // 2) t1 = h0 @ W1a  (K=512, N=2048, 16-way split-K => 2048 waves)
    gemm_wmma_kernel<LATENT, HID, LATENT / 16>
        <<<dim3(HID / 64, 16), gblk, 0, stream>>>(h0, W1a, t1);
    {
        int n = N_NODES * HID;
        bias_act_kernel<<<(n + 255) / 256, 256, 0, stream>>>(t1, b1a, HID, 1);
    }
    // 3) h1 = relu(t1 @ W1b + b1b)  (K=2048, N=2048, 16-way split-K)
    gemm_wmma_kernel<HID, HID, HID / 16>
        <<<dim3(HID / 64, 16), gblk, 0, stream>>>(t1, W1b, h1);
    {
        int n = N_NODES * HID;
        bias_act_kernel<<<(n + 255) / 256, 256, 0, stream>>>(h1, b1b, HID, 1);
    }
    // 4) agg2: h1agg = h1 + segsum(h1)
    {
        int n = N_NODES * HID;
        gin_agg_kernel<<<(n + 255) / 256, 256, 0, stream>>>(h1, eidx, h1agg, HID);
    }
    // 5) t2 = relu(h1agg @ W2a + b2a)  (K=2048, N=4096, 8-way split-K)
    gemm_wmma_kernel<HID, N_T, HID / 8>
        <<<dim3(N_T / 64, 8), gblk, 0, stream>>>(h1agg, W2a, t2);
    {
        int n = N_NODES * N_T;
        bias_act_kernel<<<(n + 255) / 256, 256, 0, stream>>>(t2, b2a, N_T, 1);
    }
    // 6) recon = t2 @ W2b + b2b  (K=4096, N=4096, 8-way split-K), accumulate into d_out+1
    gemm_wmma_kernel<N_T, N_T, N_T / 8>
        <<<dim3(N_T / 64, 8), gblk, 0, stream>>>(t2, W2b, recon);
    {
        int n = N_NODES * N_T;
        bias_act_kernel<<<(n + 255) / 256, 256, 0, stream>>>(recon, b2b, N_T, 0);
    }
    // 7) y = flat @ Wc1   (77824 rows, 512 cols; 304 blocks x 256 rows)
    {
        int rows = N_NODES * N_T;
        int rpb = 256;
        matvec_kernel<<<(rows + rpb - 1) / rpb, 512, 0, stream>>>(recon, Wc1, yv, rows, rpb);
    }
    // 8) dem_pred = sigmoid((y + bc1) . Wc2 + bc2)
    final_kernel<<<1, 512, 0, stream>>>(yv, bc1, Wc2, bc2, out);
}